// SCRFDWithPost_23691039605073
// MI455X (gfx1250) — compile-verified
//
#include <hip/hip_runtime.h>
#include <hip/hip_bf16.h>

// ---------------------------------------------------------------------------
// SCRFD post-process for MI455X (gfx1250, wave32).
// Pipeline: decode -> rank-sort -> WMMA/TDM tiled IoU bitmask (upper tri)
//           -> sequential bitset NMS scan -> masked outputs.
// Mask matrix: 8400 rows x 525 col-tiles x u16  (8.8 MB vs 282 MB reference).
// ---------------------------------------------------------------------------

#define N_ANCH   8400
#define TILES16  525          // 8400 / 16 exactly
#define PITCH16  528          // u16 per row, padded (u32 view: 264 words)
#define WORDS32  264
#define SCORE_T  0.5f
#define NMS_THR  0.4f

typedef float v2f  __attribute__((ext_vector_type(2)));
typedef float v8f  __attribute__((ext_vector_type(8)));
typedef unsigned int v4u __attribute__((ext_vector_type(4)));
typedef int   v4i  __attribute__((ext_vector_type(4)));
typedef int   v8i  __attribute__((ext_vector_type(8)));

// ---------------- Tensor Data Mover: 256B tile (64 f32) -> LDS -------------
__device__ __forceinline__ void tdm_load_64f32(unsigned lds_addr, const void* gptr) {
  unsigned long long ga = (unsigned long long)gptr;
  v4u g0;
  g0.x = 1u;                                   // count=1, is_restore=0, gather off
  g0.y = lds_addr;                             // lds_addr bytes
  g0.z = (unsigned)ga;                         // global_addr[31:0]
  g0.w = (unsigned)((ga >> 32) & 0x01FFFFFFull) | 0x80000000u; // addr[56:32], type=2
  v8i g1;
  g1[0] = (int)(2u << 16);    // data_size = 4B; no multicast / barrier / pad
  g1[1] = (int)(64u << 16);   // tensor_dim0 = 64 (low 16 in bits [31:16])
  g1[2] = (int)(1u << 16);    // tensor_dim0 hi = 0 ; tensor_dim1 = 1
  g1[3] = (int)(64u << 16);   // tensor_dim1 hi = 0 ; tile_dim0 = 64
  g1[4] = 1;                  // tile_dim1 = 1 ; tile_dim2 = 0
  g1[5] = 64;                 // tensor_dim0_stride = 64
  g1[6] = 0;
  g1[7] = 0;
  v4i z4 = {0, 0, 0, 0};
#if __has_include(<hip/amd_detail/amd_gfx1250_TDM.h>)
  v8i z8 = {0, 0, 0, 0, 0, 0, 0, 0};
  __builtin_amdgcn_tensor_load_to_lds(g0, g1, z4, z4, z8, 0);
#else
  __builtin_amdgcn_tensor_load_to_lds(g0, g1, z4, z4, 0);
#endif
}

// ---------------- 1) decode + sort keys ------------------------------------
__global__ void decode_kernel(const float* __restrict__ s0, const float* __restrict__ b0,
                              const float* __restrict__ k0, const float* __restrict__ s1,
                              const float* __restrict__ b1, const float* __restrict__ k1,
                              const float* __restrict__ s2, const float* __restrict__ b2,
                              const float* __restrict__ k2,
                              float* __restrict__ decS, float* __restrict__ decB,
                              float* __restrict__ decK, unsigned long long* __restrict__ keys) {
  int i = blockIdx.x * blockDim.x + threadIdx.x;
  if (i >= N_ANCH) return;
  const float *sp, *bp, *kp;
  int local, fw;
  float st;
  if (i < 6400)      { sp = s0; bp = b0; kp = k0; local = i;        fw = 80; st = 8.f;  }
  else if (i < 8000) { sp = s1; bp = b1; kp = k1; local = i - 6400; fw = 40; st = 16.f; }
  else               { sp = s2; bp = b2; kp = k2; local = i - 8000; fw = 20; st = 32.f; }
  float ax = (float)(local % fw) * st;
  float ay = (float)(local / fw) * st;
  float s  = 1.0f / (1.0f + expf(-sp[local]));
  decS[i] = s;
  decB[i * 4 + 0] = ax - bp[local * 4 + 0] * st;
  decB[i * 4 + 1] = ay - bp[local * 4 + 1] * st;
  decB[i * 4 + 2] = ax + bp[local * 4 + 2] * st;
  decB[i * 4 + 3] = ay + bp[local * 4 + 3] * st;
#pragma unroll
  for (int t = 0; t < 5; ++t) {
    decK[i * 10 + 2 * t + 0] = ax + kp[local * 10 + 2 * t + 0] * st;
    decK[i * 10 + 2 * t + 1] = ay + kp[local * 10 + 2 * t + 1] * st;
  }
  // monotonic float->uint flip; descending score, ascending index tiebreak
  unsigned u = __float_as_uint(s);
  u = (u & 0x80000000u) ? ~u : (u | 0x80000000u);
  keys[i] = ((unsigned long long)(~u) << 32) | (unsigned)i;
}

// ---------------- 2) rank sort (one pass, LDS-tiled) -----------------------
__global__ __launch_bounds__(256) void rank_kernel(const unsigned long long* __restrict__ keys,
                                                   unsigned* __restrict__ perm) {
  __shared__ unsigned long long tile[1024];
  int i = blockIdx.x * 256 + threadIdx.x;
  unsigned long long my = (i < N_ANCH) ? keys[i] : 0ull;
  int rank = 0;
  for (int base = 0; base < N_ANCH; base += 1024) {
    int cnt = min(1024, N_ANCH - base);
    __syncthreads();
    for (int t = threadIdx.x; t < cnt; t += 256) tile[t] = keys[base + t];
    __syncthreads();
    if (i < N_ANCH) {
#pragma unroll 4
      for (int t = 0; t < cnt; ++t) rank += (tile[t] < my) ? 1 : 0;
    }
  }
  if (i < N_ANCH) perm[rank] = (unsigned)i;
}

// ---------------- 3) gather into sorted order ------------------------------
__global__ void gather_kernel(const unsigned* __restrict__ perm, const float* __restrict__ decS,
                              const float* __restrict__ decB, const float* __restrict__ decK,
                              float* __restrict__ sS, float* __restrict__ sB,
                              float* __restrict__ sK, int* __restrict__ valid) {
  int r = blockIdx.x * blockDim.x + threadIdx.x;
  if (r >= N_ANCH) return;
  int idx = (int)perm[r];
  float s = decS[idx];
  sS[r] = s;
  valid[r] = (s > SCORE_T) ? 1 : 0;
#pragma unroll
  for (int c = 0; c < 4; ++c) sB[r * 4 + c] = decB[idx * 4 + c];
#pragma unroll
  for (int c = 0; c < 10; ++c) sK[r * 10 + c] = decK[idx * 10 + c];
}

// ---------------- 4) tiled IoU>thr bitmask (upper triangle) ----------------
// One wave per 16x16 tile. TDM loads both box tiles to LDS; WMMA f32 16x16x4
// computes the full area_i + area_j outer-sum; VALU does intersection with
// rcp-based division (TRANS op, co-executes); __ballot packs two 16-bit mask
// rows per accumulator VGPR (low16 = M=v, high16 = M=v+8).
__global__ __launch_bounds__(32) void mask_kernel(const float* __restrict__ sBoxes,
                                                  unsigned short* __restrict__ mask) {
  int tj = blockIdx.x, ti = blockIdx.y;
  if (tj < ti) return;
  __shared__ float tiles[2][16][4];   // [0] = row tile (ti), [1] = col tile (tj)
  int lane = threadIdx.x;

  unsigned ldsBase = (unsigned)(unsigned long long)(&tiles[0][0][0]);
  tdm_load_64f32(ldsBase,       (const void*)(sBoxes + (size_t)ti * 64));
  tdm_load_64f32(ldsBase + 256, (const void*)(sBoxes + (size_t)tj * 64));
  __builtin_amdgcn_s_wait_tensorcnt(0);
  __syncthreads();

  int l16 = lane & 15;
  bool hi = lane >= 16;
  float rx1 = tiles[0][l16][0], ry1 = tiles[0][l16][1];
  float rx2 = tiles[0][l16][2], ry2 = tiles[0][l16][3];
  float cx1 = tiles[1][l16][0], cy1 = tiles[1][l16][1];
  float cx2 = tiles[1][l16][2], cy2 = tiles[1][l16][3];
  float areaR = (rx2 - rx1) * (ry2 - ry1);
  float areaC = (cx2 - cx1) * (cy2 - cy1);

  // A (16x4): col0 = area_i, col1 = 1 ; B (4x16): row0 = 1, row1 = area_j
  v2f A, B;
  v8f C = {};
  A.x = hi ? 0.f : areaR;  A.y = hi ? 0.f : 1.f;
  B.x = hi ? 0.f : 1.f;    B.y = hi ? 0.f : areaC;
  v8f D = __builtin_amdgcn_wmma_f32_16x16x4_f32(false, A, false, B, (short)0, C,
                                                false, false);
  int gj = tj * 16 + l16;
#pragma unroll
  for (int v = 0; v < 8; ++v) {
    int m = v + (hi ? 8 : 0);
    float bx1 = tiles[0][m][0], by1 = tiles[0][m][1];
    float bx2 = tiles[0][m][2], by2 = tiles[0][m][3];
    float ltx = fmaxf(bx1, cx1), lty = fmaxf(by1, cy1);
    float rbx = fminf(bx2, cx2), rby = fminf(by2, cy2);
    float w = fmaxf(rbx - ltx, 0.f), h = fmaxf(rby - lty, 0.f);
    float inter = w * h;
    // D[v] = area_m + area_n ; fast reciprocal (v_rcp_f32) instead of IEEE div
    float iou = inter * __builtin_amdgcn_rcpf(D[v] - inter + 1e-9f);
    int gi = ti * 16 + m;
    bool p = (gj > gi) && (iou > NMS_THR);
    unsigned bal = (unsigned)__ballot(p);
    if (lane == 0) {
      mask[(size_t)(ti * 16 + v)     * PITCH16 + tj] = (unsigned short)(bal & 0xFFFFu);
      mask[(size_t)(ti * 16 + v + 8) * PITCH16 + tj] = (unsigned short)(bal >> 16);
    }
  }
}

// ---------------- 5) sequential greedy NMS via LDS bitset ------------------
__global__ __launch_bounds__(288) void nms_reduce(const unsigned short* __restrict__ mask,
                                                  const int* __restrict__ valid,
                                                  int* __restrict__ keep) {
  __shared__ unsigned rem[WORDS32];
  __shared__ int cur;
  int t = threadIdx.x;
  if (t < WORDS32) rem[t] = 0u;
  __syncthreads();
  for (int i = 0; i < N_ANCH; ++i) {
    if (t == 0) {
      int sup = (rem[i >> 5] >> (i & 31)) & 1;
      int k = (valid[i] != 0) && !sup;
      cur = k;
      keep[i] = k;
    }
    __syncthreads();
    if (cur && t < WORDS32) {
      const unsigned* row = (const unsigned*)(mask + (size_t)i * PITCH16);
      rem[t] |= row[t];
    }
    __syncthreads();
  }
}

// ---------------- 6) masked outputs ----------------------------------------
__global__ void finalize_kernel(const float* __restrict__ sS, const float* __restrict__ sB,
                                const float* __restrict__ sK, const int* __restrict__ keep,
                                float* __restrict__ out) {
  int r = blockIdx.x * blockDim.x + threadIdx.x;
  if (r >= N_ANCH) return;
  float m = keep[r] ? 1.f : 0.f;
  float* ob = out;                       // boxes  [8400*4]
  float* os = out + 33600;               // scores [8400]
  float* ok = out + 42000;               // kps    [8400*10]
  float* oq = out + 126000;              // keep   [8400]
#pragma unroll
  for (int c = 0; c < 4; ++c) ob[r * 4 + c] = sB[r * 4 + c] * m;
  os[r] = sS[r] * m;
#pragma unroll
  for (int c = 0; c < 10; ++c) ok[r * 10 + c] = sK[r * 10 + c] * m;
  oq[r] = m;
}

// ---------------------------------------------------------------------------
extern "C" void kernel_launch(void* const* d_in, const int* in_sizes, int n_in,
                              void* d_out, int out_size, void* d_ws, size_t ws_size,
                              hipStream_t stream) {
  (void)in_sizes; (void)n_in; (void)out_size; (void)ws_size;
  const float* s0 = (const float*)d_in[0];
  const float* b0 = (const float*)d_in[1];
  const float* k0 = (const float*)d_in[2];
  const float* s1 = (const float*)d_in[3];
  const float* b1 = (const float*)d_in[4];
  const float* k1 = (const float*)d_in[5];
  const float* s2 = (const float*)d_in[6];
  const float* b2 = (const float*)d_in[7];
  const float* k2 = (const float*)d_in[8];
  float* out = (float*)d_out;

  char* ws = (char*)d_ws;
  size_t off = 0;
  auto take = [&](size_t bytes) -> char* {
    char* p = ws + off;
    off += (bytes + 255) & ~(size_t)255;
    return p;
  };
  unsigned long long* keys = (unsigned long long*)take(N_ANCH * 8);
  float*    decS  = (float*)take(N_ANCH * 4);
  float*    decB  = (float*)take(N_ANCH * 16);
  float*    decK  = (float*)take(N_ANCH * 40);
  unsigned* perm  = (unsigned*)take(N_ANCH * 4);
  float*    sS    = (float*)take(N_ANCH * 4);
  float*    sB    = (float*)take(N_ANCH * 16);
  float*    sK    = (float*)take(N_ANCH * 40);
  int*      valid = (int*)take(N_ANCH * 4);
  int*      keep  = (int*)take(N_ANCH * 4);
  size_t maskBytes = (size_t)N_ANCH * PITCH16 * sizeof(unsigned short);
  unsigned short* mask = (unsigned short*)take(maskBytes);

  hipMemsetAsync(mask, 0, maskBytes, stream);   // zero lower triangle + pad cols

  int blocks = (N_ANCH + 255) / 256;
  decode_kernel<<<blocks, 256, 0, stream>>>(s0, b0, k0, s1, b1, k1, s2, b2, k2,
                                            decS, decB, decK, keys);
  rank_kernel<<<blocks, 256, 0, stream>>>(keys, perm);
  gather_kernel<<<blocks, 256, 0, stream>>>(perm, decS, decB, decK, sS, sB, sK, valid);
  mask_kernel<<<dim3(TILES16, TILES16), 32, 0, stream>>>(sB, mask);
  nms_reduce<<<1, 288, 0, stream>>>(mask, valid, keep);
  finalize_kernel<<<blocks, 256, 0, stream>>>(sS, sB, sK, keep, out);
}